// SAN1d_74174085202151
// MI455X (gfx1250) — compile-verified
//
#include <hip/hip_runtime.h>
#include <hip/hip_bf16.h>

// ---------------------------------------------------------------------------
// SAN1d: 4x (conv1d_same -> top-k-by-|.| -> conv1d_same) summed.
// B=64 rows, L=262144, kernel sizes {65,33,17,9}, k = L/ks.
// fp32 end-to-end using V_WMMA_F32_16X16X4_F32 (banded-Toeplitz conv-as-GEMM).
// Compile-time-specialized per kernel size so all K-step loops fully unroll.
// ---------------------------------------------------------------------------

#define L_LEN   262144
#define NROWS   64
#define CHUNK   2048          // outputs per block (8 waves x 256)
#define NK      4
#define NBINS   4096          // 12-bit |float| histogram
#define HIST_U32 (NROWS * NK * NBINS)   // 1,048,576 u32 = 4 MB

// Packed zero-padded band weights in LDS:
//   region i: [OFF, OFF+15) zeros | [OFF+15, OFF+15+ks) weights | 15 zeros
// sizes: 65+30=95, 33+30=63, 17+30=47, 9+30=39  -> total 244 floats
#define WPAD_SZ 244

typedef float v2f __attribute__((ext_vector_type(2)));
typedef float v8f __attribute__((ext_vector_type(8)));

static __device__ __forceinline__ v8f wmma4(v2f a, v2f b, v8f c) {
    // 8 args: (neg_a, A, neg_b, B, c_mod, C, reuse_a, reuse_b)
    return __builtin_amdgcn_wmma_f32_16x16x4_f32(false, a, false, b, (short)0, c,
                                                 false, false);
}

static __device__ __forceinline__ v8f v8f_zero() {
    v8f z = {0.f, 0.f, 0.f, 0.f, 0.f, 0.f, 0.f, 0.f};
    return z;
}

// Fill the padded band-weight LDS buffer (single pass, no ordering hazard).
static __device__ __forceinline__ void fill_wpad(
    float* wpad, int tid,
    const float* __restrict__ w0, const float* __restrict__ w1,
    const float* __restrict__ w2, const float* __restrict__ w3)
{
    for (int i = tid; i < WPAD_SZ; i += 256) {
        float v = 0.0f;
        if (i < 95)       { int j = i - 15;  if (j >= 0 && j < 65) v = w0[j]; }
        else if (i < 158) { int j = i - 110; if (j >= 0 && j < 33) v = w1[j]; }
        else if (i < 205) { int j = i - 173; if (j >= 0 && j < 17) v = w2[j]; }
        else              { int j = i - 220; if (j >= 0 && j < 9)  v = w3[j]; }
        wpad[i] = v;
    }
}

// Banded-Toeplitz A fragments for one kernel size, held in registers.
// A[m,t] = w[t-m]; lane holds m = lane&15, K-half = lane>>4.
template <int KS>
struct BandFrag {
    static constexpr int STEPS = (KS + 15) / 4;   // KS == 1 (mod 4) -> exact
    v2f a[STEPS];

    // wk must point at wpad + WOFF + 15 - m  (so wk[t] == band(t - m))
    __device__ __forceinline__ void load(const float* wk, int half) {
        #pragma unroll
        for (int s = 0; s < STEPS; ++s) {
            const int ta = 4 * s + 2 * half;
            a[s].x = wk[ta];
            a[s].y = wk[ta + 1];
        }
    }

    // src points at &buf[base + 16*n - pad]; base/pads even -> 8B aligned
    __device__ __forceinline__ v8f conv(const float* src, int half, v8f acc) const {
        #pragma unroll
        for (int s = 0; s < STEPS; ++s) {
            const int ta = 4 * s + 2 * half;
            v2f bf = *(const v2f*)(src + ta);
            acc = wmma4(a[s], bf, acc);
        }
        return acc;
    }
};

// ---------------------------------------------------------------------------
// Kernel 0: zero the global histogram (workspace is not re-poisoned per call)
// ---------------------------------------------------------------------------
__global__ __launch_bounds__(256) void zero_hist_kernel(uint4* g, int n4) {
    int i = blockIdx.x * 256 + threadIdx.x;
    if (i < n4) g[i] = make_uint4(0u, 0u, 0u, 0u);
}

// ---------------------------------------------------------------------------
// Kernel 1: conv via WMMA + |sim| histogram per (row, kernel)
// ---------------------------------------------------------------------------
template <int KS, int WOFF>
static __device__ __forceinline__ void hist_one(
    const float* xbuf, const float* wpad, unsigned* hist,
    int wv, int n, int half)
{
    BandFrag<KS> bfrag;
    bfrag.load(wpad + WOFF + 15 - n, half);
    v8f acc = v8f_zero();
    acc = bfrag.conv(xbuf + 32 + 256 * wv + 16 * n - (KS / 2), half, acc);
    #pragma unroll
    for (int r = 0; r < 8; ++r) {
        unsigned u = __float_as_uint(acc[r]) & 0x7fffffffu;
        atomicAdd(&hist[u >> 19], 1u);
    }
}

__global__ __launch_bounds__(256) void pass1_hist_kernel(
    const float* __restrict__ x,
    const float* __restrict__ w0, const float* __restrict__ w1,
    const float* __restrict__ w2, const float* __restrict__ w3,
    unsigned* __restrict__ ghist)
{
    __shared__ __align__(16) float    xbuf[2112];   // [c0-32, c0+2080)
    __shared__ __align__(16) float    wpad[WPAD_SZ];
    __shared__               unsigned hist[NBINS];

    const int row  = blockIdx.y;
    const int c0   = blockIdx.x * CHUNK;
    const int tid  = threadIdx.x;
    const int lane = tid & 31;
    const int wv   = tid >> 5;
    const int n    = lane & 15;
    const int half = lane >> 4;

    fill_wpad(wpad, tid, w0, w1, w2, w3);
    for (int i = tid; i < NBINS; i += 256) hist[i] = 0u;

    const float* xr = x + (size_t)row * L_LEN;
    for (int i = tid; i < 2112; i += 256) {
        int g = c0 - 32 + i;
        xbuf[i] = (g >= 0 && g < L_LEN) ? xr[g] : 0.0f;
    }
    __syncthreads();

    #define FLUSH(KI)                                                        \
        {                                                                    \
            __syncthreads();                                                 \
            unsigned* gh = ghist + ((size_t)(row * NK + (KI)) << 12);        \
            for (int i = tid; i < NBINS; i += 256) {                         \
                unsigned c = hist[i];                                        \
                if (c) atomicAdd(&gh[i], c);                                 \
                hist[i] = 0u;                                                \
            }                                                                \
            __syncthreads();                                                 \
        }

    hist_one<65,   0>(xbuf, wpad, hist, wv, n, half); FLUSH(0)
    hist_one<33,  95>(xbuf, wpad, hist, wv, n, half); FLUSH(1)
    hist_one<17, 158>(xbuf, wpad, hist, wv, n, half); FLUSH(2)
    hist_one< 9, 205>(xbuf, wpad, hist, wv, n, half); FLUSH(3)
    #undef FLUSH
}

// ---------------------------------------------------------------------------
// Kernel 2: suffix-scan the histogram, emit threshold value per (row, kernel)
// ---------------------------------------------------------------------------
__global__ __launch_bounds__(256) void pass2_thresh_kernel(
    const unsigned* __restrict__ ghist, float* __restrict__ thresh)
{
    __shared__ unsigned seg[256];
    __shared__ unsigned suff[256];

    const int bi  = blockIdx.x;          // row*4 + ki
    const int tid = threadIdx.x;
    const unsigned* h = ghist + ((size_t)bi << 12);

    unsigned loc[16];
    unsigned s = 0;
    #pragma unroll
    for (int j = 0; j < 16; ++j) { loc[j] = h[tid * 16 + j]; s += loc[j]; }
    seg[tid] = s;
    __syncthreads();

    if (tid == 0) {
        unsigned run = 0;
        for (int t = 255; t >= 0; --t) { suff[t] = run; run += seg[t]; }
    }
    __syncthreads();

    constexpr int kA[NK] = {4032, 7943, 15420, 29127};   // L // ks
    const unsigned k = (unsigned)kA[bi & 3];
    const unsigned incoming = suff[tid];
    if (incoming < k && incoming + seg[tid] >= k) {
        unsigned run   = incoming;
        int      tstar = tid * 16;
        for (int j = 15; j >= 0; --j) {
            run += loc[j];
            if (run >= k) { tstar = tid * 16 + j; break; }
        }
        thresh[bi] = __uint_as_float((unsigned)tstar << 19); // bin lower bound
    }
}

// ---------------------------------------------------------------------------
// Kernel 3: fused conv -> threshold -> conv, accumulated over 4 kernels
// ---------------------------------------------------------------------------
template <int KS, int WOFF>
static __device__ __forceinline__ void recon_one(
    const float* xbuf, float* simbuf, const float* wpad, float th,
    int c0, int tid, int wv, int n, int half, v8f& outacc)
{
    BandFrag<KS> bfrag;
    bfrag.load(wpad + WOFF + 15 - n, half);

    // ---- stage 1: similarity over extended region (9 tiles of 256) ----
    for (int tau = wv; tau < 9; tau += 8) {              // wave-uniform
        v8f acc = v8f_zero();
        acc = bfrag.conv(xbuf + 32 + 256 * tau + 16 * n - (KS / 2), half, acc);
        const int sbase = 256 * tau + 16 * n + 8 * half; // M = r + 8*half, N = n
        #pragma unroll
        for (int r = 0; r < 8; ++r) simbuf[sbase + r] = acc[r];
    }
    __syncthreads();

    // ---- threshold + zero outside [0, L) (activations are zero-padded) ----
    for (int e = tid; e < 2304; e += 256) {
        const int   g = c0 - 128 + e;
        const float v = simbuf[e];
        const bool  keep = (g >= 0) && (g < L_LEN) && (fabsf(v) >= th);
        simbuf[e] = keep ? v : 0.0f;
    }
    __syncthreads();

    // ---- stage 2: reconstruction conv of sparse activations ----
    outacc = bfrag.conv(simbuf + 128 + 256 * wv + 16 * n - (KS / 2), half, outacc);
    __syncthreads();   // simbuf reused by next kernel size
}

__global__ __launch_bounds__(256) void pass3_recon_kernel(
    const float* __restrict__ x,
    const float* __restrict__ w0, const float* __restrict__ w1,
    const float* __restrict__ w2, const float* __restrict__ w3,
    const float* __restrict__ thresh, float* __restrict__ out)
{
    __shared__ __align__(16) float xbuf[2368];     // [c0-160, c0+2208)
    __shared__ __align__(16) float simbuf[2304];   // [c0-128, c0+2176)
    __shared__ __align__(16) float wpad[WPAD_SZ];

    const int row  = blockIdx.y;
    const int c0   = blockIdx.x * CHUNK;
    const int tid  = threadIdx.x;
    const int lane = tid & 31;
    const int wv   = tid >> 5;
    const int n    = lane & 15;
    const int half = lane >> 4;

    fill_wpad(wpad, tid, w0, w1, w2, w3);

    const float* xr = x + (size_t)row * L_LEN;
    for (int i = tid; i < 2368; i += 256) {
        int g = c0 - 160 + i;
        xbuf[i] = (g >= 0 && g < L_LEN) ? xr[g] : 0.0f;
    }
    __syncthreads();

    v8f outacc = v8f_zero();
    const float* thr = thresh + row * NK;

    recon_one<65,   0>(xbuf, simbuf, wpad, thr[0], c0, tid, wv, n, half, outacc);
    recon_one<33,  95>(xbuf, simbuf, wpad, thr[1], c0, tid, wv, n, half, outacc);
    recon_one<17, 158>(xbuf, simbuf, wpad, thr[2], c0, tid, wv, n, half, outacc);
    recon_one< 9, 205>(xbuf, simbuf, wpad, thr[3], c0, tid, wv, n, half, outacc);

    float* orow = out + (size_t)row * L_LEN + c0 + 256 * wv + 16 * n + 8 * half;
    float4 lo = make_float4(outacc[0], outacc[1], outacc[2], outacc[3]);
    float4 hi = make_float4(outacc[4], outacc[5], outacc[6], outacc[7]);
    ((float4*)orow)[0] = lo;
    ((float4*)orow)[1] = hi;
}

// ---------------------------------------------------------------------------
// Host launch
// ---------------------------------------------------------------------------
extern "C" void kernel_launch(void* const* d_in, const int* in_sizes, int n_in,
                              void* d_out, int out_size, void* d_ws, size_t ws_size,
                              hipStream_t stream) {
    const float* x  = (const float*)d_in[0];
    const float* w0 = (const float*)d_in[1];
    const float* w1 = (const float*)d_in[2];
    const float* w2 = (const float*)d_in[3];
    const float* w3 = (const float*)d_in[4];
    float* out = (float*)d_out;

    unsigned* ghist  = (unsigned*)d_ws;                                   // 4 MB
    float*    thresh = (float*)((char*)d_ws + (size_t)HIST_U32 * 4);      // 1 KB

    // 0) zero global histogram
    {
        int n4 = HIST_U32 / 4;
        zero_hist_kernel<<<(n4 + 255) / 256, 256, 0, stream>>>((uint4*)ghist, n4);
    }
    // 1) conv + histogram
    {
        dim3 grid(L_LEN / CHUNK, NROWS);
        pass1_hist_kernel<<<grid, 256, 0, stream>>>(x, w0, w1, w2, w3, ghist);
    }
    // 2) thresholds
    pass2_thresh_kernel<<<NROWS * NK, 256, 0, stream>>>(ghist, thresh);
    // 3) fused conv -> sparsify -> conv -> sum
    {
        dim3 grid(L_LEN / CHUNK, NROWS);
        pass3_recon_kernel<<<grid, 256, 0, stream>>>(x, w0, w1, w2, w3, thresh, out);
    }
}